// ConvCNP_54881092108481
// MI455X (gfx1250) — compile-verified
//
#include <hip/hip_runtime.h>
#include <hip/hip_bf16.h>
#include <math.h>

#define BATCH 8
#define NC    2048
#define NTT   2048   // num_t == T == 2048

typedef __attribute__((ext_vector_type(2))) float v2f;
typedef __attribute__((ext_vector_type(4))) float v4f;
typedef __attribute__((ext_vector_type(8))) float v8f;

// Bare v_exp_f32 (arg in log2 domain). Safe here: RBF args are <= 0
// (underflow flushes to 0, which is what a Gaussian tail should do) and
// sigmoid/softplus args behave correctly at +/-inf.
__device__ __forceinline__ float fast_exp2(float x) {
#if __has_builtin(__builtin_amdgcn_exp2f)
    return __builtin_amdgcn_exp2f(x);
#else
    return exp2f(x);
#endif
}
#define LOG2E 1.44269504088896f

// ---------------- Kernel A: global min/max over xc and xt -------------------
__global__ void minmax_kernel(const float* __restrict__ xc,
                              const float* __restrict__ xt,
                              float* __restrict__ bounds) {
    __shared__ float smin[1024];
    __shared__ float smax[1024];
    int tid = threadIdx.x;
    float mn = 3.0e38f, mx = -3.0e38f;
    for (int i = tid; i < BATCH * NC; i += blockDim.x) {
        float a = xc[i]; mn = fminf(mn, a); mx = fmaxf(mx, a);
        float b = xt[i]; mn = fminf(mn, b); mx = fmaxf(mx, b);
    }
    smin[tid] = mn; smax[tid] = mx;
    __syncthreads();
    for (int s = blockDim.x >> 1; s > 0; s >>= 1) {
        if (tid < s) {
            smin[tid] = fminf(smin[tid], smin[tid + s]);
            smax[tid] = fmaxf(smax[tid], smax[tid + s]);
        }
        __syncthreads();
    }
    if (tid == 0) { bounds[0] = smin[0]; bounds[1] = smax[0]; }
}

// ------- Kernel B: fused Ktc@phi (RBF generated inline) + sigmoid MLP -------
// out: hbuf (B, 8, NT) for the conv stack
__global__ void hsmooth_kernel(const float* __restrict__ xc,
                               const float* __restrict__ yc,
                               const float* __restrict__ psi_ls_p,
                               const float* __restrict__ psi_os_p,
                               const float* __restrict__ mlp_w,   // (3,8)
                               const float* __restrict__ mlp_b,   // (8)
                               const float* __restrict__ bounds,
                               float* __restrict__ hbuf) {
    int gid = blockIdx.x * blockDim.x + threadIdx.x;   // B*NT threads
    int b = gid / NTT, i = gid % NTT;
    float lower = bounds[0], upper = bounds[1];
    float ti = lower + (upper - lower) * ((float)i / (float)(NTT - 1));
    float ls = psi_ls_p[0], os = psi_os_p[0];
    float coef = -0.5f / (ls * ls) * LOG2E;            // exp(x) = exp2(x*log2e)
    const float* xcb = xc + b * NC;
    const float* ycb = yc + b * NC;
    float s0 = 0.0f, s1 = 0.0f;
    #pragma unroll 4
    for (int j = 0; j < NC; ++j) {
        float d = ti - xcb[j];
        float e = fast_exp2(coef * d * d);
        s0 += e;
        s1 += e * ycb[j];
    }
    s0 *= os; s1 *= os;
    float h0 = s0;
    float h1 = s1 / (s0 + 1e-8f);
    #pragma unroll
    for (int c = 0; c < 8; ++c) {
        float z = ti * mlp_w[c] + h0 * mlp_w[8 + c] + h1 * mlp_w[16 + c] + mlp_b[c];
        float e = fast_exp2(-z * LOG2E);
        hbuf[(b * 8 + c) * NTT + i] = 1.0f / (1.0f + e);
    }
}

// --------------- Kernel C: conv1d k=5 pad=2 + ReLU (templated) --------------
template <int CI, int CO>
__global__ void conv5_relu_kernel(const float* __restrict__ x,   // (B,CI,NT)
                                  const float* __restrict__ w,   // (CO,CI,5)
                                  const float* __restrict__ bias,
                                  float* __restrict__ y) {       // (B,CO,NT)
    int gid = blockIdx.x * blockDim.x + threadIdx.x;   // B*CO*NT
    int t = gid % NTT;
    int rest = gid / NTT;
    int o = rest % CO;
    int b = rest / CO;
    const float* xb = x + (size_t)b * CI * NTT;
    const float* wo = w + o * CI * 5;
    float acc = bias[o];
    #pragma unroll
    for (int k = 0; k < 5; ++k) {
        int tt = t + k - 2;
        if (tt >= 0 && tt < NTT) {
            #pragma unroll
            for (int ci = 0; ci < CI; ++ci)
                acc += xb[ci * NTT + tt] * wo[ci * 5 + k];
        }
    }
    y[((size_t)b * CO + o) * NTT + t] = fmaxf(acc, 0.0f);
}

// --------- Kernel Z: stream zeros into the (B,T,T) var output region --------
__global__ void zerofill_kernel(v4f* __restrict__ p, long n4) {
    long i = (long)blockIdx.x * blockDim.x + threadIdx.x;
    long stride = (long)gridDim.x * blockDim.x;
    v4f z = {0.f, 0.f, 0.f, 0.f};
    for (; i < n4; i += stride)
        __builtin_nontemporal_store(z, p + i);         // streaming b128 stores
}

// ----- Kernel D: Krho @ f via V_WMMA_F32_16X16X4_F32, then mean/var head ----
// One wave per 16-row output tile; 8 waves/block; 16 tile-groups per batch.
__global__ void __launch_bounds__(256)
final_wmma_kernel(const float* __restrict__ xt,
                  const float* __restrict__ c4,       // (B,8,NT)  conv output
                  const float* __restrict__ rho_ls_p,
                  const float* __restrict__ rho_os_p,
                  const float* __restrict__ mean_w,   // (8,1)
                  const float* __restrict__ mean_b,   // (1)
                  const float* __restrict__ var_w,    // (8,1)
                  const float* __restrict__ var_b,    // (1)
                  const float* __restrict__ bounds,
                  float* __restrict__ mean_out,       // (B,T)
                  float* __restrict__ var_out) {      // (B,T,T), pre-zeroed
    __shared__ float ldsF[NTT * 8];                   // f staged as [j][n], n<8
    int b  = blockIdx.x >> 4;
    int tg = blockIdx.x & 15;
    int tid = threadIdx.x;

    const float* fb = c4 + (size_t)b * 8 * NTT;
    for (int idx = tid; idx < NTT * 8; idx += 256) {
        int j = idx >> 3, n = idx & 7;
        ldsF[idx] = fb[n * NTT + j];                  // transpose (8,NT)->(NT,8)
    }
    __syncthreads();

    int wave = tid >> 5;
    int lane = tid & 31;
    int n    = lane & 15;           // N index (channel); cols 8..15 are zero pad
    int nb   = n & 7;               // clamped LDS column (masked to zero below)
    int hi   = (lane >= 16);
    int i0   = (tg * 8 + wave) * 16;
    float maskf = (n < 8) ? 1.0f : 0.0f;   // hoisted: no EXEC games in the loop

    float lower = bounds[0], upper = bounds[1];
    float step  = (upper - lower) / (float)(NTT - 1);
    float ls = rho_ls_p[0], os = rho_os_p[0];
    float coef = -0.5f / (ls * ls) * LOG2E;
    float xti  = xt[b * NTT + i0 + n];                // A row for this lane
    int   koff = hi ? 2 : 0;

    v8f acc = {};
    #pragma unroll 2
    for (int j0 = 0; j0 < NTT; j0 += 4) {
        int jA = j0 + koff;
        float tA = lower + step * (float)jA;
        float tB = tA + step;
        float dA = xti - tA, dB = xti - tB;
        // A fragment: Krho[i, jA], Krho[i, jA+1] generated in registers
        v2f a;
        a.x = os * fast_exp2(coef * dA * dA);
        a.y = os * fast_exp2(coef * dB * dB);
        // B fragment: f[jA, n], f[jA+1, n]; unconditional loads + zero mask
        v2f bb;
        bb.x = ldsF[jA * 8 + nb] * maskf;
        bb.y = ldsF[jA * 8 + 8 + nb] * maskf;
        acc = __builtin_amdgcn_wmma_f32_16x16x4_f32(
                  false, a, false, bb, (short)0, acc, false, false);
    }

    // Head: per row i -> mean = f.mean_w + mean_b ; var_ii = softplus(f.var_w + var_b)
    float mw = (n < 8) ? mean_w[n] : 0.0f;
    float vw = (n < 8) ? var_w[n] : 0.0f;
    float mb = mean_b[0], vb = var_b[0];
    #pragma unroll
    for (int r = 0; r < 8; ++r) {
        float pm = acc[r] * mw;
        float pv = acc[r] * vw;
        #pragma unroll
        for (int m = 1; m <= 8; m <<= 1) {        // reduce within 16-lane N group
            pm += __shfl_xor(pm, m, 32);
            pv += __shfl_xor(pv, m, 32);
        }
        if (n == 0) {
            int row = i0 + r + (hi ? 8 : 0);      // C/D layout: hi half = rows+8
            mean_out[b * NTT + row] = pm + mb;
            float x = pv + vb;
            float e = fast_exp2(-fabsf(x) * LOG2E);
            float sp = fmaxf(x, 0.0f) + log1pf(e);  // stable softplus
            var_out[((size_t)b * NTT + row) * NTT + row] = sp;
        }
    }
}

extern "C" void kernel_launch(void* const* d_in, const int* in_sizes, int n_in,
                              void* d_out, int out_size, void* d_ws, size_t ws_size,
                              hipStream_t stream) {
    const float* xc     = (const float*)d_in[0];
    const float* yc     = (const float*)d_in[1];
    const float* xt     = (const float*)d_in[2];
    // d_in[3] = num_t (int, == 2048, hardcoded)
    const float* psi_ls = (const float*)d_in[4];
    const float* psi_os = (const float*)d_in[5];
    const float* rho_ls = (const float*)d_in[6];
    const float* rho_os = (const float*)d_in[7];
    const float* mlp_w  = (const float*)d_in[8];
    const float* mlp_b  = (const float*)d_in[9];
    const float* w1 = (const float*)d_in[10]; const float* b1 = (const float*)d_in[11];
    const float* w2 = (const float*)d_in[12]; const float* b2 = (const float*)d_in[13];
    const float* w3 = (const float*)d_in[14]; const float* b3 = (const float*)d_in[15];
    const float* w4 = (const float*)d_in[16]; const float* b4 = (const float*)d_in[17];
    const float* mean_w = (const float*)d_in[18]; const float* mean_b = (const float*)d_in[19];
    const float* var_w  = (const float*)d_in[20]; const float* var_b  = (const float*)d_in[21];

    float* ws     = (float*)d_ws;
    float* bounds = ws;                               // 2 floats (padded to 16)
    float* hbuf   = ws + 16;                          // (8, 8,2048)  = 131072
    float* c1     = hbuf + 8 * 8  * 2048;             // (8,16,2048)  = 262144
    float* c2     = c1   + 8 * 16 * 2048;             // (8,32,2048)  = 524288
    float* c3     = c2   + 8 * 32 * 2048;             // (8,16,2048)  = 262144
    float* c4     = c3   + 8 * 16 * 2048;             // (8, 8,2048)  = 131072

    float* mean_out = (float*)d_out;                  // (8,2048)
    float* var_out  = mean_out + 8 * 2048;            // (8,2048,2048)

    minmax_kernel<<<1, 1024, 0, stream>>>(xc, xt, bounds);
    hsmooth_kernel<<<(8 * 2048) / 256, 256, 0, stream>>>(
        xc, yc, psi_ls, psi_os, mlp_w, mlp_b, bounds, hbuf);
    conv5_relu_kernel<8, 16><<<(8 * 16 * 2048) / 256, 256, 0, stream>>>(hbuf, w1, b1, c1);
    conv5_relu_kernel<16, 32><<<(8 * 32 * 2048) / 256, 256, 0, stream>>>(c1, w2, b2, c2);
    conv5_relu_kernel<32, 16><<<(8 * 16 * 2048) / 256, 256, 0, stream>>>(c2, w3, b3, c3);
    conv5_relu_kernel<16, 8><<<(8 * 8 * 2048) / 256, 256, 0, stream>>>(c3, w4, b4, c4);
    zerofill_kernel<<<4096, 256, 0, stream>>>(
        (v4f*)var_out, (long)8 * 2048 * 2048 / 4);
    final_wmma_kernel<<<128, 256, 0, stream>>>(
        xt, c4, rho_ls, rho_os, mean_w, mean_b, var_w, var_b, bounds,
        mean_out, var_out);
}